// AttentionDecoder_27427661152742
// MI455X (gfx1250) — compile-verified
//
#include <hip/hip_runtime.h>
#include <math.h>

// Problem constants (from reference setup_inputs)
constexpr int B  = 32;
constexpr int L  = 2048;
constexpr int T  = 64;
constexpr int De = 512;
constexpr int Dd = 256;
constexpr int NS = 256;          // N_STATES
constexpr int Din = De + Dd;     // 768
constexpr int G3  = 3 * NS;      // 768

typedef __attribute__((ext_vector_type(2))) float v2f;
typedef __attribute__((ext_vector_type(8))) float v8f;

// Truncate a generic pointer to a __shared__ object to its 32-bit LDS byte
// offset (flat LDS addresses carry the offset in addr[31:0]).
__device__ __forceinline__ unsigned lds_off(const void* p) {
    return (unsigned)(unsigned long long)(uintptr_t)p;
}

// ---------------------------------------------------------------------------
// keys = enc(M x K) @ Wk(K x N), row-major f32, V_WMMA_F32_16X16X4_F32.
//
// Block: 256 threads = 8 waves arranged 2(M) x 4(N); block tile 32x64,
// K chunked by 32. A (32x32) and B (32x64) tiles are staged to LDS with
// CDNA5 async global->LDS b128 copies (ASYNCcnt), then WMMA fragments are
// read from LDS per the ISA 7.12.2 f32 layouts:
//   A 16x4 : lane r=lane&15 is row M; lane-half selects K-pair {0,1}/{2,3}
//   B 4x16 : VGPR v holds K=v (lanes 0-15) / K=v+2 (lanes 16-31), N=lane
//   C 16x16: VGPR v holds M=v (lanes 0-15) / M=v+8 (lanes 16-31), N=lane
// ---------------------------------------------------------------------------
constexpr int PA = 36;   // LDS pitch of A tile (words): 144B, 16B-aligned, bank-spread
constexpr int PB = 68;   // LDS pitch of B tile (words): 272B, 16B-aligned

__global__ void keys_gemm_wmma(const float* __restrict__ A,
                               const float* __restrict__ Bm,
                               float* __restrict__ C,
                               int M, int K, int N) {
    __shared__ float sA[32 * PA];     // 4.6 KB
    __shared__ float sB[32 * PB];     // 8.7 KB

    const int tid  = threadIdx.x;
    const int wave = tid >> 5;
    const int lane = tid & 31;
    const int r    = lane & 15;
    const int hi   = lane >> 4;
    const int wm   = wave >> 2;            // 0..1  -> M sub-tile
    const int wn   = wave & 3;             // 0..3  -> N sub-tile
    const int mB   = blockIdx.y * 32;      // block M origin
    const int nB   = blockIdx.x * 64;      // block N origin

    // --- async staging thread mapping -------------------------------------
    // A tile: 32 rows x 32 f32 = 4 KB = 256 lanes x 16B  (8 lanes per row)
    const int arow = tid >> 3;
    const int acol = (tid & 7) * 4;
    // B tile: 32 rows x 64 f32 = 8 KB = 256 lanes x 2 x 16B (16 lanes per row)
    const int brow = tid >> 4;
    const int bcol = (tid & 15) * 4;

    const unsigned long long baseA = (unsigned long long)(A + (size_t)mB * K);
    const unsigned long long baseB = (unsigned long long)(Bm + nB);
    const unsigned ldsA  = lds_off(&sA[arow * PA + acol]);
    const unsigned ldsB0 = lds_off(&sB[brow * PB + bcol]);
    const unsigned ldsB1 = lds_off(&sB[(brow + 16) * PB + bcol]);

    v8f acc = {};
    for (int k0 = 0; k0 < K; k0 += 32) {
        __syncthreads();                              // previous chunk consumed
        // ---- issue async global->LDS tile copies (ASYNCcnt) ----
        const unsigned offA  = (unsigned)((arow * K + k0 + acol) * 4);
        const unsigned offB0 = (unsigned)(((k0 + brow) * N + bcol) * 4);
        const unsigned offB1 = (unsigned)(((k0 + brow + 16) * N + bcol) * 4);
        asm volatile("global_load_async_to_lds_b128 %0, %1, %2"
                     :: "v"(ldsA), "v"(offA), "s"(baseA) : "memory");
        asm volatile("global_load_async_to_lds_b128 %0, %1, %2"
                     :: "v"(ldsB0), "v"(offB0), "s"(baseB) : "memory");
        asm volatile("global_load_async_to_lds_b128 %0, %1, %2"
                     :: "v"(ldsB1), "v"(offB1), "s"(baseB) : "memory");
        asm volatile("s_wait_asynccnt 0x0" ::: "memory");
        __syncthreads();                              // all waves' tiles in LDS

        // ---- 8 x WMMA over the 32-deep K chunk ----
#pragma unroll
        for (int kk = 0; kk < 32; kk += 4) {
            const int ka = kk + 2 * hi;
            v2f a, b;
            a.x = sA[(wm * 16 + r) * PA + ka + 0];
            a.y = sA[(wm * 16 + r) * PA + ka + 1];
            b.x = sB[(ka + 0) * PB + wn * 16 + r];
            b.y = sB[(ka + 1) * PB + wn * 16 + r];
            acc = __builtin_amdgcn_wmma_f32_16x16x4_f32(false, a, false, b,
                                                        (short)0, acc, false, false);
        }
    }

    float* Crow = C + (size_t)(mB + wm * 16 + 8 * hi) * N + nB + wn * 16 + r;
#pragma unroll
    for (int v = 0; v < 8; ++v) Crow[(size_t)v * N] = acc[v];
}

// ---------------------------------------------------------------------------
// h := 0
// ---------------------------------------------------------------------------
__global__ void init_h(float* __restrict__ h) {
    h[blockIdx.x * NS + threadIdx.x] = 0.0f;
}

// ---------------------------------------------------------------------------
// q[b,n] = bq[n] + sum_k h[b,k] * Wq[k,n]      (grid=B, block=NS)
// ---------------------------------------------------------------------------
__global__ void q_kernel(const float* __restrict__ h,
                         const float* __restrict__ Wq,
                         const float* __restrict__ bq,
                         float* __restrict__ q) {
    const int b = blockIdx.x, n = threadIdx.x;
    __shared__ float hs[NS];
    hs[n] = h[b * NS + n];
    __syncthreads();
    float acc = bq[n];
#pragma unroll 4
    for (int k = 0; k < NS; ++k) acc += hs[k] * Wq[k * NS + n];
    q[b * NS + n] = acc;
}

// ---------------------------------------------------------------------------
// score[b,l] = sum_n tanh(keys[b,l,n] + q[b,n]) * Ws[n]  - (1-mask)*2^-31
// Wave-per-row: one wave32 reduces one l row; loads are 512B-contiguous
// float4 per instruction; 5-step shfl_xor butterfly reduction.
// grid=(L/64, B), block=256 (8 waves, 8 rows each).
// ---------------------------------------------------------------------------
__global__ void score_kernel(const float* __restrict__ keys,
                             const float* __restrict__ q,
                             const unsigned char* __restrict__ enc_mask,
                             const float* __restrict__ Ws,
                             float* __restrict__ score) {
    const int b    = blockIdx.y;
    const int wave = threadIdx.x >> 5;
    const int lane = threadIdx.x & 31;
    __shared__ float qs[NS];
    __shared__ float ws[NS];
    qs[threadIdx.x] = q[b * NS + threadIdx.x];
    ws[threadIdx.x] = Ws[threadIdx.x];
    __syncthreads();

    const int l0 = blockIdx.x * 64 + wave * 8;
#pragma unroll
    for (int i = 0; i < 8; ++i) {
        const int l = l0 + i;
        const float* krow = keys + ((size_t)b * L + l) * NS;
        float s = 0.0f;
#pragma unroll
        for (int c = 0; c < 2; ++c) {
            const int n = c * 128 + lane * 4;
            float4 kv = *(const float4*)(krow + n);
            s += tanhf(kv.x + qs[n + 0]) * ws[n + 0];
            s += tanhf(kv.y + qs[n + 1]) * ws[n + 1];
            s += tanhf(kv.z + qs[n + 2]) * ws[n + 2];
            s += tanhf(kv.w + qs[n + 3]) * ws[n + 3];
        }
#pragma unroll
        for (int off = 16; off > 0; off >>= 1) s += __shfl_xor(s, off, 32);
        if (lane == 0) {
            const float madd =
                (1.0f - (enc_mask[b * L + l] ? 1.0f : 0.0f)) * 0x1p-31f;
            score[(size_t)b * L + l] = s - madd;
        }
    }
}

// ---------------------------------------------------------------------------
// In-place softmax over L for each b (grid=B, block=256, 8 elems/thread).
// ---------------------------------------------------------------------------
__global__ void softmax_kernel(float* __restrict__ score) {
    const int b = blockIdx.x, tid = threadIdx.x;
    float* s = score + (size_t)b * L;
    __shared__ float red[256];

    float v[8];
    float m = -INFINITY;
#pragma unroll
    for (int i = 0; i < 8; ++i) { v[i] = s[tid + i * 256]; m = fmaxf(m, v[i]); }
    red[tid] = m;
    __syncthreads();
    for (int off = 128; off > 0; off >>= 1) {
        if (tid < off) red[tid] = fmaxf(red[tid], red[tid + off]);
        __syncthreads();
    }
    m = red[0];
    __syncthreads();

    float sum = 0.0f;
#pragma unroll
    for (int i = 0; i < 8; ++i) { v[i] = __expf(v[i] - m); sum += v[i]; }
    red[tid] = sum;
    __syncthreads();
    for (int off = 128; off > 0; off >>= 1) {
        if (tid < off) red[tid] += red[tid + off];
        __syncthreads();
    }
    const float inv = 1.0f / red[0];
#pragma unroll
    for (int i = 0; i < 8; ++i) s[tid + i * 256] = v[i] * inv;
}

// ---------------------------------------------------------------------------
// glimpse[b,de] = sum_l attn[b,l] * enc[b,l,de]   grid=(De/256, B), block=256
// attn staged through LDS; enc access coalesced across threads per l.
// ---------------------------------------------------------------------------
__global__ void glimpse_kernel(const float* __restrict__ attn,
                               const float* __restrict__ enc,
                               float* __restrict__ glimpse) {
    const int b  = blockIdx.y;
    const int de = blockIdx.x * blockDim.x + threadIdx.x;
    __shared__ float a[L];                      // 8 KB of the 320 KB LDS
    for (int i = threadIdx.x; i < L; i += blockDim.x) a[i] = attn[(size_t)b * L + i];
    __syncthreads();

    const float* erow = enc + (size_t)b * L * De + de;
    float g = 0.0f;
    for (int l = 0; l < L; ++l) {
        __builtin_prefetch(erow + (size_t)(l + 32) * De, 0, 0);   // global_prefetch_b8
        g += a[l] * erow[(size_t)l * De];
    }
    glimpse[b * De + de] = g;
}

// ---------------------------------------------------------------------------
// xg(32x768) = [glimpse | x_t](32x768) @ gru_kernel(768x768) + bias0
// f32 WMMA, one wave per 16x16 tile. grid=(G3/16, 2), block=32.
// ---------------------------------------------------------------------------
__global__ void gru_xg_gemm(const float* __restrict__ glimpse,
                            const float* __restrict__ dec,     // (B,T,Dd)
                            const float* __restrict__ Wg,      // (Din,G3)
                            const float* __restrict__ bias0,
                            float* __restrict__ xg, int t) {
    const int lane = threadIdx.x & 31;
    const int r    = lane & 15;
    const int hi   = lane >> 4;
    const int n0   = blockIdx.x * 16;
    const int m    = blockIdx.y * 16 + r;      // batch row

    const float* grow = glimpse + m * De;
    const float* drow = dec + ((size_t)m * T + t) * Dd;

    v8f acc = {};
    for (int k = 0; k < Din; k += 4) {
        const int ka = k + 2 * hi;             // ka,ka+1 never straddle De (mult of 4)
        v2f a, b;
        if (ka < De) { a.x = grow[ka]; a.y = grow[ka + 1]; }
        else         { a.x = drow[ka - De]; a.y = drow[ka + 1 - De]; }
        b.x = Wg[(size_t)(ka + 0) * G3 + n0 + r];
        b.y = Wg[(size_t)(ka + 1) * G3 + n0 + r];
        acc = __builtin_amdgcn_wmma_f32_16x16x4_f32(false, a, false, b,
                                                    (short)0, acc, false, false);
    }
    const float bn = bias0[n0 + r];
    float* Crow = xg + (size_t)(blockIdx.y * 16 + 8 * hi) * G3 + n0 + r;
#pragma unroll
    for (int v = 0; v < 8; ++v) Crow[(size_t)v * G3] = acc[v] + bn;
}

// ---------------------------------------------------------------------------
// rg(32x768) = h(32x256) @ gru_rec_kernel(256x768) + bias1
// ---------------------------------------------------------------------------
__global__ void gru_rg_gemm(const float* __restrict__ h,
                            const float* __restrict__ Wr,      // (NS,G3)
                            const float* __restrict__ bias1,
                            float* __restrict__ rg) {
    const int lane = threadIdx.x & 31;
    const int r    = lane & 15;
    const int hi   = lane >> 4;
    const int n0   = blockIdx.x * 16;
    const float* Arow = h + (size_t)(blockIdx.y * 16 + r) * NS;

    v8f acc = {};
    for (int k = 0; k < NS; k += 4) {
        const int ka = k + 2 * hi;
        v2f a, b;
        a.x = Arow[ka + 0];
        a.y = Arow[ka + 1];
        b.x = Wr[(size_t)(ka + 0) * G3 + n0 + r];
        b.y = Wr[(size_t)(ka + 1) * G3 + n0 + r];
        acc = __builtin_amdgcn_wmma_f32_16x16x4_f32(false, a, false, b,
                                                    (short)0, acc, false, false);
    }
    const float bn = bias1[n0 + r];
    float* Crow = rg + (size_t)(blockIdx.y * 16 + 8 * hi) * G3 + n0 + r;
#pragma unroll
    for (int v = 0; v < 8; ++v) Crow[(size_t)v * G3] = acc[v] + bn;
}

// ---------------------------------------------------------------------------
// GRU gates + state update + output write. grid=B, block=NS.
// ---------------------------------------------------------------------------
__global__ void gru_update(const float* __restrict__ xg,
                           const float* __restrict__ rg,
                           const unsigned char* __restrict__ dmask,
                           float* __restrict__ h,
                           float* __restrict__ out, int t) {
    const int b = blockIdx.x, n = threadIdx.x;
    const float* xb = xg + (size_t)b * G3;
    const float* rb = rg + (size_t)b * G3;

    const float z  = 1.0f / (1.0f + __expf(-(xb[n] + rb[n])));
    const float rr = 1.0f / (1.0f + __expf(-(xb[NS + n] + rb[NS + n])));
    const float hh = tanhf(xb[2 * NS + n] + rr * rb[2 * NS + n]);
    const float hp = h[b * NS + n];
    const float hn = z * hp + (1.0f - z) * hh;
    const float hnext = (dmask[b * T + t] != 0) ? hn : hp;
    h[b * NS + n] = hnext;
    out[((size_t)b * T + t) * NS + n] = hnext;
}

// ---------------------------------------------------------------------------
// Host-side orchestration
// ---------------------------------------------------------------------------
extern "C" void kernel_launch(void* const* d_in, const int* in_sizes, int n_in,
                              void* d_out, int out_size, void* d_ws, size_t ws_size,
                              hipStream_t stream) {
    (void)in_sizes; (void)n_in; (void)out_size; (void)ws_size;

    const float*         enc    = (const float*)d_in[0];          // (B,L,De)
    const float*         dec    = (const float*)d_in[1];          // (B,T,Dd)
    const unsigned char* emask  = (const unsigned char*)d_in[2];  // (B,L) bool
    const unsigned char* dmask  = (const unsigned char*)d_in[3];  // (B,T) bool
    const float*         Wk     = (const float*)d_in[4];          // (De,NS)
    const float*         Wq     = (const float*)d_in[5];          // (NS,NS)
    const float*         bq     = (const float*)d_in[6];          // (NS,)
    const float*         Ws     = (const float*)d_in[7];          // (NS,1)
    const float*         Wg     = (const float*)d_in[8];          // (Din,G3)
    const float*         Wr     = (const float*)d_in[9];          // (NS,G3)
    const float*         gbias  = (const float*)d_in[10];         // (2,G3)
    float*               out    = (float*)d_out;                  // (B,T,NS)

    // Workspace partition (floats)
    float* ws      = (float*)d_ws;
    float* keys    = ws;                               // B*L*NS   = 16.78M
    float* score   = keys    + (size_t)B * L * NS;     // B*L      = 65536
    float* q       = score   + (size_t)B * L;          // B*NS
    float* glimpse = q       + (size_t)B * NS;         // B*De
    float* xg      = glimpse + (size_t)B * De;         // B*G3
    float* rg      = xg      + (size_t)B * G3;         // B*G3
    float* h       = rg      + (size_t)B * G3;         // B*NS

    // h = 0
    init_h<<<B, NS, 0, stream>>>(h);

    // keys = enc @ Wk   (M=B*L, K=De, N=NS), WMMA f32 + async LDS staging
    {
        dim3 grid(NS / 64, (B * L) / 32);
        keys_gemm_wmma<<<grid, 256, 0, stream>>>(enc, Wk, keys, B * L, De, NS);
    }

    for (int t = 0; t < T; ++t) {
        q_kernel<<<B, NS, 0, stream>>>(h, Wq, bq, q);
        score_kernel<<<dim3(L / 64, B), 256, 0, stream>>>(keys, q, emask, Ws, score);
        softmax_kernel<<<B, 256, 0, stream>>>(score);
        glimpse_kernel<<<dim3(De / 256, B), 256, 0, stream>>>(score, enc, glimpse);
        gru_xg_gemm<<<dim3(G3 / 16, B / 16), 32, 0, stream>>>(glimpse, dec, Wg,
                                                              gbias, xg, t);
        gru_rg_gemm<<<dim3(G3 / 16, B / 16), 32, 0, stream>>>(h, Wr, gbias + G3, rg);
        gru_update<<<B, NS, 0, stream>>>(xg, rg, dmask, h, out, t);
    }
}